// KeyedLayer_3564822855839
// MI455X (gfx1250) — compile-verified
//
#include <hip/hip_runtime.h>
#include <stdint.h>

#ifndef __has_builtin
#define __has_builtin(x) 0
#endif

#if __has_builtin(__builtin_amdgcn_global_load_async_to_lds_b32)
#define HAVE_ASYNC_LDS 1
#else
#define HAVE_ASYNC_LDS 0
#endif

// ---- probe kernel, FIRST in file (never launched): verify fp32 atomic ------
// Want to see exactly one global_atomic_add_f32 in the asm snippet window.
__global__ void atomic_probe(float* p, float v) {
  unsafeAtomicAdd(p + threadIdx.x, v);
}

// Address-space-qualified pointer typedefs (AS1 = global, AS3 = LDS)
typedef __attribute__((address_space(1))) int gas_int;
typedef __attribute__((address_space(3))) int las_int;

// ---- CDNA5 async global->LDS helpers (gfx1250) ------------------------------
__device__ __forceinline__ void async_g2l_b32(const void* gptr, void* lptr) {
#if HAVE_ASYNC_LDS
  __builtin_amdgcn_global_load_async_to_lds_b32(
      (gas_int*)(gptr), (las_int*)(lptr), 0, 0);
#else
  (void)gptr; (void)lptr;
#endif
}

__device__ __forceinline__ void wait_async0() {
#if HAVE_ASYNC_LDS
#if __has_builtin(__builtin_amdgcn_s_wait_asynccnt)
  __builtin_amdgcn_s_wait_asynccnt(0);
#else
  asm volatile("s_wait_asynccnt 0" ::: "memory");
#endif
#endif
}

#define SCAT_TILE 512

// ---- hot kernel: full 512-nnz tiles, branch-free ---------------------------
// yT[row,:] += xT[col,:] * v   (xT/yT are [N,64]; 1 wave per nnz, float2/lane)
__global__ __launch_bounds__(256) void scatter_coo_full(const int* __restrict__ rows,
                                                        const int* __restrict__ cols,
                                                        const float* __restrict__ vals,
                                                        const float* __restrict__ xT,
                                                        float* __restrict__ yT) {
  __shared__ int   s_rows[SCAT_TILE];
  __shared__ int   s_cols[SCAT_TILE];
  __shared__ float s_vals[SCAT_TILE];

  const int base = blockIdx.x * SCAT_TILE;
  const int i0 = threadIdx.x, i1 = threadIdx.x + 256;

#if HAVE_ASYNC_LDS
  async_g2l_b32(rows + base + i0, &s_rows[i0]);
  async_g2l_b32(rows + base + i1, &s_rows[i1]);
  async_g2l_b32(cols + base + i0, &s_cols[i0]);
  async_g2l_b32(cols + base + i1, &s_cols[i1]);
  async_g2l_b32(vals + base + i0, &s_vals[i0]);
  async_g2l_b32(vals + base + i1, &s_vals[i1]);
  wait_async0();
#else
  s_rows[i0] = rows[base + i0]; s_rows[i1] = rows[base + i1];
  s_cols[i0] = cols[base + i0]; s_cols[i1] = cols[base + i1];
  s_vals[i0] = vals[base + i0]; s_vals[i1] = vals[base + i1];
#endif
  __syncthreads();

  const int warp = threadIdx.x >> 5;
  const int lane = threadIdx.x & 31;
  const float2* __restrict__ xT2 = (const float2*)xT + lane;  // lane-fixed base
  float* __restrict__ dstl = yT + lane * 2;

  #pragma unroll 2
  for (int i = warp; i < SCAT_TILE; i += 8) {
    const int   col = s_cols[i];
    const int   row = s_rows[i];
    const float v   = s_vals[i];
    const float2 xv = xT2[(size_t)col * 32];      // 256B contiguous per wave
    float* dst = dstl + (size_t)row * 64;
    unsafeAtomicAdd(dst,     xv.x * v);           // global_atomic_add_f32
    unsafeAtomicAdd(dst + 1, xv.y * v);
  }
}

// ---- tail kernel: nonzeros in [Kfull, K), one wave per nnz -----------------
__global__ __launch_bounds__(256) void scatter_tail(const int* __restrict__ rows,
                                                    const int* __restrict__ cols,
                                                    const float* __restrict__ vals,
                                                    const float* __restrict__ xT,
                                                    float* __restrict__ yT,
                                                    int Kfull, int K) {
  const int k = Kfull + (int)((blockIdx.x * blockDim.x + threadIdx.x) >> 5);
  const int lane = threadIdx.x & 31;
  if (k >= K) return;
  const int col = cols[k], row = rows[k];
  const float v = vals[k];
  const float2 xv = ((const float2*)(xT + (size_t)col * 64))[lane];
  float* dst = yT + (size_t)row * 64 + lane * 2;
  unsafeAtomicAdd(dst,     xv.x * v);
  unsafeAtomicAdd(dst + 1, xv.y * v);
}

// ---- zero a float4-aligned region ------------------------------------------
__global__ __launch_bounds__(256) void zero_f4(float4* __restrict__ p, long n4) {
  long i = (long)blockIdx.x * blockDim.x + threadIdx.x;
  long stride = (long)gridDim.x * blockDim.x;
  float4 z = {0.f, 0.f, 0.f, 0.f};
  for (; i < n4; i += stride) p[i] = z;
}

// ---- transpose x [64, N] -> xT [N, 64] -------------------------------------
__global__ __launch_bounds__(256) void transpose_b2n(const float* __restrict__ x,
                                                     float* __restrict__ xT, int N) {
  __shared__ float tile[64][65];  // pad to 65: conflict-free on 64 banks
  const int n0 = blockIdx.x * 64;
  const int tx = threadIdx.x & 63;  // n within tile
  const int tq = threadIdx.x >> 6;  // 0..3
  const bool full = (n0 + 64 <= N);

  if (full) {
#if HAVE_ASYNC_LDS
    #pragma unroll
    for (int b = tq; b < 64; b += 4)
      async_g2l_b32(x + (size_t)b * N + (n0 + tx), &tile[b][tx]);
    wait_async0();
#else
    #pragma unroll
    for (int b = tq; b < 64; b += 4)
      tile[b][tx] = x[(size_t)b * N + (n0 + tx)];
#endif
  } else {
    int n = n0 + tx;
    if (n < N)
      for (int b = tq; b < 64; b += 4)
        tile[b][tx] = x[(size_t)b * N + n];
  }
  __syncthreads();

  const int b = threadIdx.x & 63;  // b within tile (coalesced writes)
  for (int nl = tq; nl < 64; nl += 4) {
    int nn = n0 + nl;
    if (nn < N) xT[(size_t)nn * 64 + b] = tile[b][nl];
  }
}

// ---- transpose yT [N, 64] -> y [64, N] -------------------------------------
__global__ __launch_bounds__(256) void transpose_n2b(const float* __restrict__ yT,
                                                     float* __restrict__ y, int N) {
  __shared__ float tile[64][65];
  const int n0 = blockIdx.x * 64;
  const int b  = threadIdx.x & 63;
  const int tq = threadIdx.x >> 6;

  for (int nl = tq; nl < 64; nl += 4) {
    int nn = n0 + nl;
    if (nn < N) tile[nl][b] = yT[(size_t)nn * 64 + b];  // coalesced over b
  }
  __syncthreads();

  const int tx = threadIdx.x & 63;
  const int n  = n0 + tx;
  if (n < N) {
    for (int bb = tq; bb < 64; bb += 4)
      y[(size_t)bb * N + n] = tile[tx][bb];             // coalesced over n
  }
}

// ---- fallback: direct scatter into [B,N] output (if ws too small) ----------
__global__ __launch_bounds__(256) void scatter_direct(const int* __restrict__ rows,
                                                      const int* __restrict__ cols,
                                                      const float* __restrict__ vals,
                                                      const float* __restrict__ x,
                                                      float* __restrict__ y,
                                                      int K, int N) {
  const int k = (int)(((long)blockIdx.x * blockDim.x + threadIdx.x) >> 5);
  const int lane = threadIdx.x & 31;
  if (k >= K) return;
  const int col = cols[k], row = rows[k];
  const float v = vals[k];
  #pragma unroll
  for (int b = lane; b < 64; b += 32) {
    float xv = x[(size_t)b * N + col];
    unsafeAtomicAdd(&y[(size_t)b * N + row], xv * v);
  }
}

extern "C" void kernel_launch(void* const* d_in, const int* in_sizes, int n_in,
                              void* d_out, int out_size, void* d_ws, size_t ws_size,
                              hipStream_t stream) {
  const float* x    = (const float*)d_in[0];  // [B=64, N]
  const float* vals = (const float*)d_in[1];  // [K]
  const int*   rows = (const int*)d_in[2];    // [K]
  const int*   cols = (const int*)d_in[3];    // [K]
  float* out = (float*)d_out;                 // [B=64, N]

  const int BN = in_sizes[0];
  const int N  = BN / 64;
  const int K  = in_sizes[2];

  const size_t mat_bytes = (size_t)N * 64 * sizeof(float);
  const int nTiles = (N + 63) / 64;

  if (ws_size >= 2 * mat_bytes) {
    float* xT = (float*)d_ws;
    float* yT = (float*)((char*)d_ws + mat_bytes);

    zero_f4<<<1024, 256, 0, stream>>>((float4*)yT, (long)N * 64 / 4);
    transpose_b2n<<<nTiles, 256, 0, stream>>>(x, xT, N);

    const int Kfull = K & ~(SCAT_TILE - 1);
    if (Kfull > 0)
      scatter_coo_full<<<Kfull / SCAT_TILE, 256, 0, stream>>>(rows, cols, vals, xT, yT);
    const int tail = K - Kfull;
    if (tail > 0) {
      int tblocks = (tail * 32 + 255) / 256;
      scatter_tail<<<tblocks, 256, 0, stream>>>(rows, cols, vals, xT, yT, Kfull, K);
    }

    transpose_n2b<<<nTiles, 256, 0, stream>>>(yT, out, N);
  } else {
    zero_f4<<<1024, 256, 0, stream>>>((float4*)out, (long)BN / 4);
    long threads = (long)K * 32;
    int blocks = (int)((threads + 255) / 256);
    scatter_direct<<<blocks, 256, 0, stream>>>(rows, cols, vals, x, out, K, N);
  }
}